// ResidualBlock_20375324852526
// MI455X (gfx1250) — compile-verified
//
#include <hip/hip_runtime.h>

typedef __attribute__((ext_vector_type(2))) float v2f;
typedef __attribute__((ext_vector_type(4))) float v4f;
typedef __attribute__((ext_vector_type(8))) float v8f;

#define NPTS   60000
#define CIN0   64
#define COUT   96
#define KOFF   27
#define NTILES (NPTS / 16)   // 3750, exact
#define LDSPAD 32            // floats; shifts consecutive fragment-rows by 32 banks

// ---------------------------------------------------------------------------
// Sparse conv pass: for kernel offset k (blockIdx.x), gather 16 feature rows,
// GEMM against W[k] staged in LDS (pre-swizzled to WMMA B-fragment order),
// scatter-add 16x96 results with f32 atomics.
//
// LDS swizzle: element W[c][n] -> wlds[(c>>1)*RSTRIDE + 2n + (c&1)], so a
// B fragment (K pair 4s+2h / 4s+2h+1 at column n) is one contiguous b64.
// ---------------------------------------------------------------------------
template <int CIN>
__global__ __launch_bounds__(256) void spconv_wmma(
    const float* __restrict__ feat,     // (N+1) x CIN (only rows <N gathered)
    const float* __restrict__ W,        // K x CIN x COUT
    const int*   __restrict__ in_idx,   // K x N
    const int*   __restrict__ out_idx,  // K x N (padded entries == N)
    float*       __restrict__ outacc)   // (N+1) x COUT, zero-initialized
{
    constexpr int ROWS    = CIN / 2;            // fragment-rows (index 2s+h)
    constexpr int RSTRIDE = 2 * COUT + LDSPAD;  // 224 floats, bank shift 32
    __shared__ float wlds[ROWS * RSTRIDE];

    const int k = blockIdx.x;
    const float* wk = W + (size_t)k * CIN * COUT;
    for (int t = threadIdx.x; t < CIN * COUT; t += blockDim.x) {
        const int c = t / COUT, n = t - c * COUT;
        wlds[(c >> 1) * RSTRIDE + 2 * n + (c & 1)] = wk[t];
    }
    __syncthreads();

    const int lane  = threadIdx.x & 31;
    const int wave  = threadIdx.x >> 5;
    const int half  = lane >> 4;      // 0: lanes 0-15, 1: lanes 16-31
    const int l16   = lane & 15;
    const int kbase = k * NPTS;
    const int wpb   = blockDim.x >> 5;   // 8 waves per block

    for (int tile = blockIdx.y * wpb + wave; tile < NTILES;
         tile += gridDim.y * wpb) {
        const int i0 = tile << 4;

        // ---- per-lane index loads (one each), out rows via shuffle ----
        const int idxin  = in_idx [kbase + i0 + l16];
        const int idxout = out_idx[kbase + i0 + l16];
        int orow[8];
#pragma unroll
        for (int r = 0; r < 8; ++r)
            orow[r] = __shfl(idxout, half * 8 + r, 32);

        // ---- gather A fragments: lane = row M=l16, K pair 2*half ----
        const float* frow = feat + (size_t)idxin * CIN + 2 * half;
        v2f a[CIN / 4];
#pragma unroll
        for (int s = 0; s < CIN / 4; ++s)
            a[s] = *(const v2f*)(frow + 4 * s);

        // ---- 6 column tiles of 16 cover COUT=96 ----
#pragma unroll
        for (int j = 0; j < 6; ++j) {
            v8f c = {};
#pragma unroll
            for (int s = 0; s < CIN / 4; ++s) {
                const v2f b = *(const v2f*)&wlds[(2 * s + half) * RSTRIDE +
                                                 2 * (j * 16 + l16)];
                c = __builtin_amdgcn_wmma_f32_16x16x4_f32(
                        false, a[s], false, b, (short)0, c, false, false);
            }
            const int col = j * 16 + l16;
#pragma unroll
            for (int r = 0; r < 8; ++r)
                atomicAdd(&outacc[(size_t)orow[r] * COUT + col], c[r]);
        }
    }
}

// ---------------------------------------------------------------------------
// Fused middle stage: skip = bn_d(x @ Wd) -> d_out (scratch), and
// h1 = relu(bn1(h1)) in place. Wd staged in LDS (same swizzle), 8 waves/blk.
// ---------------------------------------------------------------------------
__global__ __launch_bounds__(256) void mid_fuse(
    const float* __restrict__ x,   const float* __restrict__ Wd,
    const float* __restrict__ g1,  const float* __restrict__ b1,
    const float* __restrict__ m1,  const float* __restrict__ v1,
    const float* __restrict__ gd,  const float* __restrict__ bd,
    const float* __restrict__ md,  const float* __restrict__ vd,
    float* __restrict__ h1, float* __restrict__ skip)
{
    constexpr int ROWS    = CIN0 / 2;
    constexpr int RSTRIDE = 2 * COUT + LDSPAD;
    __shared__ float wlds[ROWS * RSTRIDE];

    for (int t = threadIdx.x; t < CIN0 * COUT; t += blockDim.x) {
        const int c = t / COUT, n = t - c * COUT;
        wlds[(c >> 1) * RSTRIDE + 2 * n + (c & 1)] = Wd[t];
    }
    __syncthreads();   // all threads reach this before any tile guard

    const int lane = threadIdx.x & 31;
    const int wave = threadIdx.x >> 5;
    const int tile = blockIdx.x * 8 + wave;      // wave-uniform
    if (tile >= NTILES) return;                  // whole wave exits

    const int half = lane >> 4;
    const int l16  = lane & 15;
    const int i0   = tile << 4;

    // ---- skip GEMM: dense rows (i0 + l16), CIN0 = 64 ----
    const float* frow = x + (size_t)(i0 + l16) * CIN0 + 2 * half;
    v2f a[CIN0 / 4];
#pragma unroll
    for (int s = 0; s < CIN0 / 4; ++s)
        a[s] = *(const v2f*)(frow + 4 * s);

#pragma unroll
    for (int j = 0; j < 6; ++j) {
        v8f c = {};
#pragma unroll
        for (int s = 0; s < CIN0 / 4; ++s) {
            const v2f b = *(const v2f*)&wlds[(2 * s + half) * RSTRIDE +
                                             2 * (j * 16 + l16)];
            c = __builtin_amdgcn_wmma_f32_16x16x4_f32(
                    false, a[s], false, b, (short)0, c, false, false);
        }
        const int   col = j * 16 + l16;
        const float sc  = gd[col] * rsqrtf(vd[col] + 1e-5f);
        const float mc  = md[col], bc = bd[col];
#pragma unroll
        for (int r = 0; r < 8; ++r)
            skip[(size_t)(i0 + half * 8 + r) * COUT + col] =
                (c[r] - mc) * sc + bc;
    }

    // ---- h1 = relu(bn1(h1)) for these 16 rows (lanes stride columns) ----
    for (int c0 = lane; c0 < COUT; c0 += 32) {
        const float sc = g1[c0] * rsqrtf(v1[c0] + 1e-5f);
        const float mc = m1[c0], bc = b1[c0];
#pragma unroll
        for (int r = 0; r < 16; ++r) {
            const size_t idx = (size_t)(i0 + r) * COUT + c0;
            const float  v  = (h1[idx] - mc) * sc + bc;
            h1[idx] = v > 0.f ? v : 0.f;
        }
    }
}

// ---------------------------------------------------------------------------
// Final: out = relu(bn2(conv2_acc) + skip); skip resident in d_out. float4.
// ---------------------------------------------------------------------------
__global__ __launch_bounds__(256) void final_fuse(
    const float* __restrict__ h2,
    const float* __restrict__ g2, const float* __restrict__ b2,
    const float* __restrict__ m2, const float* __restrict__ v2,
    float* __restrict__ out)
{
    const int q = blockIdx.x * blockDim.x + threadIdx.x;   // float4 index
    if (q >= (NPTS * COUT) / 4) return;
    const int base = q * 4;
    const int c0   = base % COUT;                 // 96 % 4 == 0, no row wrap

    v4f acc = *(const v4f*)(h2 + base);
    v4f sk  = *(const v4f*)(out + base);
    v4f res;
#pragma unroll
    for (int e = 0; e < 4; ++e) {
        const int   c  = c0 + e;
        const float sc = g2[c] * rsqrtf(v2[c] + 1e-5f);
        const float t  = b2[c] - m2[c] * sc;
        const float v  = acc[e] * sc + t + sk[e];
        res[e] = v > 0.f ? v : 0.f;
    }
    *(v4f*)(out + base) = res;
}

// ---------------------------------------------------------------------------
extern "C" void kernel_launch(void* const* d_in, const int* in_sizes, int n_in,
                              void* d_out, int out_size, void* d_ws, size_t ws_size,
                              hipStream_t stream) {
    const float* x   = (const float*)d_in[0];
    const float* W1  = (const float*)d_in[1];
    const float* g1  = (const float*)d_in[2];
    const float* b1  = (const float*)d_in[3];
    const float* m1  = (const float*)d_in[4];
    const float* v1  = (const float*)d_in[5];
    const float* W2  = (const float*)d_in[6];
    const float* g2  = (const float*)d_in[7];
    const float* b2  = (const float*)d_in[8];
    const float* m2  = (const float*)d_in[9];
    const float* v2  = (const float*)d_in[10];
    const float* Wd  = (const float*)d_in[11];
    const float* gd  = (const float*)d_in[12];
    const float* bd  = (const float*)d_in[13];
    const float* md  = (const float*)d_in[14];
    const float* vd  = (const float*)d_in[15];
    const int* in_idx  = (const int*)d_in[16];
    const int* out_idx = (const int*)d_in[17];
    float* out = (float*)d_out;

    const size_t rows = (size_t)(NPTS + 1) * COUT;   // +1 trash row
    if (ws_size < 2 * rows * sizeof(float)) return;
    float* h1 = (float*)d_ws;
    float* h2 = h1 + rows;

    hipMemsetAsync(h1, 0, rows * sizeof(float), stream);
    hipMemsetAsync(h2, 0, rows * sizeof(float), stream);

    dim3 cgrid(KOFF, 32);   // 27 offsets x 32 tile-groups, 8 waves/block
    spconv_wmma<CIN0><<<cgrid, 256, 0, stream>>>(x, W1, in_idx, out_idx, h1);

    mid_fuse<<<(NTILES + 7) / 8, 256, 0, stream>>>(
        x, Wd, g1, b1, m1, v1, gd, bd, md, vd, h1, out);

    spconv_wmma<COUT><<<cgrid, 256, 0, stream>>>(h1, W2, in_idx, out_idx, h2);

    final_fuse<<<((NPTS * COUT) / 4 + 255) / 256, 256, 0, stream>>>(
        h2, g2, b2, m2, v2, out);
}